// FeatureSelfAttention_30374008717912
// MI455X (gfx1250) — compile-verified
//
#include <hip/hip_runtime.h>
#include <hip/hip_bf16.h>

// CDNA5 / gfx1250, wave32.
typedef __attribute__((ext_vector_type(2))) float v2f;
typedef __attribute__((ext_vector_type(8))) float v8f;

#define FD    64      // feature dim
#define FPAD  68      // padded LDS row (bank-conflict-free column access)
#define TILE  16      // timesteps per block
#define NWAVE 8       // 256 threads / wave32

__device__ __forceinline__ float wave_reduce_max(float v) {
#pragma unroll
  for (int off = 16; off > 0; off >>= 1)
    v = fmaxf(v, __shfl_xor(v, off, 32));
  return v;
}
__device__ __forceinline__ float wave_reduce_sum(float v) {
#pragma unroll
  for (int off = 16; off > 0; off >>= 1)
    v += __shfl_xor(v, off, 32);
  return v;
}

__global__ __launch_bounds__(256) void fsa_kernel(
    const float* __restrict__ x,      // (B*T, 64) row-major
    const float* __restrict__ Wq,     // (64, 64) row-major (out, in)
    const float* __restrict__ Wk,     // (64, 64)
    const float* __restrict__ gamma,  // (64,)
    float* __restrict__ out,          // (B*T, 64)
    float* __restrict__ attn)         // (B*T, 64, 64)
{
  __shared__ float xs[TILE][FPAD];
  __shared__ float qs[TILE][FPAD];
  __shared__ float ks[TILE][FPAD];

  const int  tid  = threadIdx.x;
  const int  lane = tid & 31;
  const int  wave = tid >> 5;
  const long pos0 = (long)blockIdx.x * TILE;

  // ---- Phase 1: stage 16x64 fp32 x-tile into LDS (4 floats / thread) ----
  {
    const int row = tid >> 4;              // (tid*4) / 64
    const int col = (tid & 15) << 2;       // (tid*4) % 64
    const float4 v = *(const float4*)(x + pos0 * FD + row * FD + col);
    *(float4*)&xs[row][col] = v;
  }
  __syncthreads();

  // ---- Phase 2: Q/K projections via V_WMMA_F32_16X16X4_F32 (exact fp32) ----
  // Q[t,p] = sum_f x[t,f] * Wq[p,f]  -> A = x-tile (16xK), B[k][n] = W[n][k].
  // Waves 0-3: Q N-chunks 0..3 ; waves 4-7: K N-chunks 0..3.
  {
    const float* W  = (wave < 4) ? Wq : Wk;
    const int n0    = (wave & 3) * 16;
    const int col   = lane & 15;           // N within chunk (B/D frag)
    const int arow  = lane & 15;           // M (A frag)
    const int koff  = (lane >> 4) << 1;    // lanes 0-15: K=0,1 ; 16-31: K=2,3

    v8f acc = {};
#pragma unroll
    for (int kb = 0; kb < 16; ++kb) {      // K = 64 in steps of 4
      const int kbase = kb * 4 + koff;
      v2f a, b;
      a.x = xs[arow][kbase];
      a.y = xs[arow][kbase + 1];
      const float* wp = W + (n0 + col) * FD + kbase;
      b.x = wp[0];
      b.y = wp[1];
      acc = __builtin_amdgcn_wmma_f32_16x16x4_f32(
          /*neg_a=*/false, a, /*neg_b=*/false, b,
          /*c_mod=*/(short)0, acc, /*reuse_a=*/false, /*reuse_b=*/false);
    }

    // D layout: VGPR r -> M=r (lanes 0-15) / M=r+8 (lanes 16-31), N=lane&15.
    float (*dst)[FPAD] = (wave < 4) ? qs : ks;
    const int mbase = (lane >> 4) << 3;
#pragma unroll
    for (int r = 0; r < 8; ++r)
      dst[mbase + r][n0 + col] = acc[r];
  }
  __syncthreads();

  // ---- Phase 3: per-row softmax of the QK outer product, NT-stream attn ----
  // rows = 16 positions x 64 feature-rows; one row per wave per iteration.
  const float inv_scale = 0.125f;          // 1/sqrt(64)
  for (int r = wave; r < TILE * FD; r += NWAVE) {
    const int p = r >> 6;                  // position within tile
    const int i = r & 63;                  // feature row
    const float q  = qs[p][i] * inv_scale; // broadcast LDS read
    const float k0 = ks[p][lane];
    const float k1 = ks[p][lane + 32];
    const float s0 = q * k0;
    const float s1 = q * k1;

    const float m  = wave_reduce_max(fmaxf(s0, s1));
    const float e0 = __expf(s0 - m);
    const float e1 = __expf(s1 - m);
    const float x0 = xs[p][lane];
    const float x1 = xs[p][lane + 32];
    const float Z  = wave_reduce_sum(e0 + e1);
    const float wx = wave_reduce_sum(e0 * x0 + e1 * x1);
    const float iz = 1.0f / Z;

    // attn row: 64 contiguous fp32 -> two coalesced 128B NT stores.
    float* ap = attn + (((pos0 + p) * FD + i) << 6);
    __builtin_nontemporal_store(e0 * iz, ap + lane);
    __builtin_nontemporal_store(e1 * iz, ap + lane + 32);

    if (lane == 0) {
      const float mixed = wx * iz;         // sum_j attn*x
      out[(pos0 + p) * FD + i] = xs[p][i] + gamma[i] * mixed;
    }
  }
}

extern "C" void kernel_launch(void* const* d_in, const int* in_sizes, int n_in,
                              void* d_out, int out_size, void* d_ws, size_t ws_size,
                              hipStream_t stream) {
  const float* x     = (const float*)d_in[0];
  const float* Wq    = (const float*)d_in[1];
  const float* Wk    = (const float*)d_in[2];
  const float* gamma = (const float*)d_in[3];

  float* out = (float*)d_out;
  const long n_x       = in_sizes[0];        // B*T*F
  const long positions = n_x / FD;           // B*T
  float* attn = out + n_x;                   // tuple order: (out, attn_weights)

  dim3 grid((unsigned)(positions / TILE)), block(256);
  hipLaunchKernelGGL(fsa_kernel, grid, block, 0, stream,
                     x, Wq, Wk, gamma, out, attn);
}